// PostProcess_66082366816770
// MI455X (gfx1250) — compile-verified
//
#include <hip/hip_runtime.h>
#include <math.h>

typedef __attribute__((ext_vector_type(2))) float v2f;
typedef __attribute__((ext_vector_type(8))) float v8f;

// Problem constants (match reference): B=32, Q=1024, NC=256, IOU_THR=0.5
#define NBATCH 32
#define NQ     1024
#define NCLS   256

// ---------------------------------------------------------------------------
// Kernel A: softmax scores + labels.
// One wave per 16 queries. Lane L handles query (L&15), class half (L>>4):
// classes [half*128, half*128+128). Softmax denominator is reduced with
// V_WMMA_F32_16X16X4_F32 (exact f32: A = per-half partial sums as 16x4,
// B = ones), result broadcast via LDS.
// ---------------------------------------------------------------------------
__global__ __launch_bounds__(256) void scores_labels_kernel(
    const float* __restrict__ logits,
    float* __restrict__ scores_out,
    float* __restrict__ labels_out)
{
    __shared__ float denLds[8][16];

    const int lane = threadIdx.x & 31;
    const int wave = threadIdx.x >> 5;
    const int m16  = lane & 15;   // query within the 16-query tile (M index)
    const int half = lane >> 4;   // which 128-class half this lane owns
    const int q    = (blockIdx.x * 8 + wave) * 16 + m16;

    const float*  p  = logits + (size_t)q * NCLS + half * 128;
    __builtin_prefetch(p, 0, 1);                // global_prefetch_b8
    const float4* p4 = (const float4*)p;

    // Pass 1: row max (all 256) + foreground max/argmax (classes 0..254).
    float mx  = -INFINITY;
    float fgv = -INFINITY;
    int   fgi = 0;
    const int cbase = half * 128;
#pragma unroll 4
    for (int i = 0; i < 32; ++i) {
        float4 v = p4[i];
        mx = fmaxf(mx, fmaxf(fmaxf(v.x, v.y), fmaxf(v.z, v.w)));
        int c = cbase + 4 * i;
        if (v.x > fgv) { fgv = v.x; fgi = c;     }
        if (v.y > fgv) { fgv = v.y; fgi = c + 1; }
        if (v.z > fgv) { fgv = v.z; fgi = c + 2; }
        if ((c + 3) < (NCLS - 1) && v.w > fgv) { fgv = v.w; fgi = c + 3; }
    }

    // Merge the two halves of each query (wave32: partner is lane^16).
    mx = fmaxf(mx, __shfl_xor(mx, 16));
    {
        float ov = __shfl_xor(fgv, 16);
        int   oi = __shfl_xor(fgi, 16);
        if (ov > fgv || (ov == fgv && oi < fgi)) { fgv = ov; fgi = oi; }
    }

    // Pass 2: per-half partial sum of exp(x - mx) (hits L2; input fits in 192MB).
    float partial = 0.0f;
#pragma unroll 4
    for (int i = 0; i < 32; ++i) {
        float4 v = p4[i];
        partial += expf(v.x - mx) + expf(v.y - mx) +
                   expf(v.z - mx) + expf(v.w - mx);
    }

    // Exact f32 reduction via WMMA: A[m][k] (16x4) holds partials at k=2*half,
    // zeros elsewhere; B = ones(4x16) => D[m][n] = denominator of query m.
    // A layout (32-bit A 16x4): lanes 0-15 -> K={0,1}, lanes 16-31 -> K={2,3}.
    v2f a; a.x = partial; a.y = 0.0f;
    v2f bones; bones.x = 1.0f; bones.y = 1.0f;
    v8f cacc = {};
    v8f dres = __builtin_amdgcn_wmma_f32_16x16x4_f32(
        /*neg_a=*/false, a, /*neg_b=*/false, bones,
        /*c_mod=*/(short)0, cacc, /*reuse_a=*/false, /*reuse_b=*/false);

    // D layout: lane 0 holds D[M=0..7][N=0], lane 16 holds D[M=8..15][N=0].
    if (m16 == 0) {
#pragma unroll
        for (int r = 0; r < 8; ++r) denLds[wave][half * 8 + r] = dres[r];
    }
    __syncthreads();
    float den = denLds[wave][m16];

    float score = expf(fgv - mx) / den;
    if (half == 0) {
        scores_out[q] = score;
        labels_out[q] = (float)fgi;
    }
}

// ---------------------------------------------------------------------------
// Kernel B: cxcywh -> xyxy, scaled by [w,h,w,h] (target_sizes = [h,w]).
// ---------------------------------------------------------------------------
__global__ __launch_bounds__(256) void boxes_kernel(
    const float* __restrict__ in_boxes,
    const int*   __restrict__ tsz,
    float* __restrict__ boxes_out)
{
    int idx = blockIdx.x * blockDim.x + threadIdx.x;   // 0..B*Q-1
    int b   = idx >> 10;
    float4 v = ((const float4*)in_boxes)[idx];
    float H = (float)tsz[2 * b];
    float W = (float)tsz[2 * b + 1];
    float4 o;
    o.x = (v.x - 0.5f * v.z) * W;
    o.y = (v.y - 0.5f * v.w) * H;
    o.z = (v.x + 0.5f * v.z) * W;
    o.w = (v.y + 0.5f * v.w) * H;
    ((float4*)boxes_out)[idx] = o;
}

// ---------------------------------------------------------------------------
// Kernel C: greedy NMS, one block (1024 threads = 32 waves) per image.
// Stable descending bitonic sort in LDS, then sequential suppression scan.
// ---------------------------------------------------------------------------
__global__ __launch_bounds__(1024) void nms_kernel(
    const float* __restrict__ scores,
    const float* __restrict__ boxes,
    float* __restrict__ keep_out)
{
    __shared__ float  s_sc[NQ];
    __shared__ int    s_id[NQ];
    __shared__ float4 s_box[NQ];
    __shared__ int    s_keep[NQ];

    const int tid = threadIdx.x;
    const int b   = blockIdx.x;

    s_sc[tid] = scores[b * NQ + tid];
    s_id[tid] = tid;
    __syncthreads();

    // Bitonic sort: descending by score, ties broken by lower original index
    // (matches stable jnp.argsort(-scores)).
    for (int k = 2; k <= NQ; k <<= 1) {
        for (int j = k >> 1; j > 0; j >>= 1) {
            int ixj = tid ^ j;
            if (ixj > tid) {
                float sa = s_sc[tid], sb = s_sc[ixj];
                int   ia = s_id[tid], ib = s_id[ixj];
                bool aFirst = (sa > sb) || (sa == sb && ia < ib);
                bool up = ((tid & k) == 0);
                if (up ? !aFirst : aFirst) {
                    s_sc[tid] = sb; s_sc[ixj] = sa;
                    s_id[tid] = ib; s_id[ixj] = ia;
                }
            }
            __syncthreads();
        }
    }

    // Gather boxes in sorted order into LDS.
    float4 bb = ((const float4*)boxes)[b * NQ + s_id[tid]];
    s_box[tid]  = bb;
    float areaJ = (bb.z - bb.x) * (bb.w - bb.y);
    s_keep[tid] = 1;
    __syncthreads();

    // Sequential greedy suppression: pivot i suppresses all later j with
    // IoU > 0.5, only if pivot itself is still kept.
    for (int i = 0; i < NQ - 1; ++i) {
        if (s_keep[i] && tid > i && s_keep[tid]) {
            float4 bi = s_box[i];
            float areaI = (bi.z - bi.x) * (bi.w - bi.y);
            float lx = fmaxf(bi.x, bb.x), ly = fmaxf(bi.y, bb.y);
            float rx = fminf(bi.z, bb.z), ry = fminf(bi.w, bb.w);
            float iw = fmaxf(rx - lx, 0.0f), ih = fmaxf(ry - ly, 0.0f);
            float inter = iw * ih;
            float uni   = areaI + areaJ - inter;
            float iou   = inter / fmaxf(uni, 1e-9f);
            if (iou > 0.5f) s_keep[tid] = 0;
        }
        __syncthreads();
    }

    // Scatter keep flags back to original query order.
    keep_out[b * NQ + s_id[tid]] = s_keep[tid] ? 1.0f : 0.0f;
}

// ---------------------------------------------------------------------------
// Launch. Output layout (flat f32): scores[32768] | labels[32768] |
// boxes[131072] | keep[32768]  => 229376 floats total.
// ---------------------------------------------------------------------------
extern "C" void kernel_launch(void* const* d_in, const int* in_sizes, int n_in,
                              void* d_out, int out_size, void* d_ws, size_t ws_size,
                              hipStream_t stream)
{
    (void)in_sizes; (void)n_in; (void)out_size; (void)d_ws; (void)ws_size;

    const float* logits = (const float*)d_in[0];   // [32,1024,256] f32
    const float* pboxes = (const float*)d_in[1];   // [32,1024,4]   f32
    const int*   tsz    = (const int*)d_in[2];     // [32,2]        i32

    float* out      = (float*)d_out;
    float* scoresO  = out;                              // 32768
    float* labelsO  = out + NBATCH * NQ;                // 32768
    float* boxesO   = out + 2 * NBATCH * NQ;            // 131072
    float* keepO    = out + 2 * NBATCH * NQ + NBATCH * NQ * 4; // 32768

    // A: 32768 queries / (8 waves * 16 queries) = 256 blocks of 256 threads.
    scores_labels_kernel<<<256, 256, 0, stream>>>(logits, scoresO, labelsO);
    // B: 32768 box threads.
    boxes_kernel<<<128, 256, 0, stream>>>(pboxes, tsz, boxesO);
    // C: one block per image.
    nms_kernel<<<NBATCH, NQ, 0, stream>>>(scoresO, boxesO, keepO);
}